// GATIDConv_36000415875688
// MI455X (gfx1250) — compile-verified
//
#include <hip/hip_runtime.h>
#include <math.h>

#define N_NODES   100000
#define N_EDGES   1600000
#define D_IN      256
#define C_OUT     64
#define NEG_SLOPE 0.2f
#define NID_CNT   10000
#define E_TOT     (N_EDGES + N_NODES)

// Transposed-W LDS layout: Wt[n][k], row stride padded to 260 dwords so the
// 32 lanes' ds_load_b64 of a B fragment hit 64 distinct LDS banks.
#define WT_STRIDE 260
#define LDS_BYTES (C_OUT * WT_STRIDE * 4)   // 66560 B (dynamic LDS; WGP has 320KB)

typedef __attribute__((ext_vector_type(2))) float v2f;
typedef __attribute__((ext_vector_type(8))) float v8f;

// ---------------------------------------------------------------------------
// float atomic max via signed/unsigned integer atomics (valid with -inf init)
// ---------------------------------------------------------------------------
__device__ __forceinline__ void atomicMaxF(float* addr, float v) {
    if (v >= 0.0f) {
        atomicMax((int*)addr, __float_as_int(v));
    } else {
        atomicMin((unsigned int*)addr, __float_as_uint(v));
    }
}

// ---------------------------------------------------------------------------
// Cooperative transpose of W [256x64] row-major -> LDS Wt[n][k] (padded rows).
// Each thread handles float4 chunks of W; 4 scalar ds stores per chunk.
// ---------------------------------------------------------------------------
__device__ __forceinline__ void load_W_transposed(const float* __restrict__ W,
                                                  float* __restrict__ Wt,
                                                  int t) {
    #pragma unroll
    for (int i = 0; i < (D_IN * C_OUT / 4) / 256; ++i) {    // 16 iterations
        const int q  = i * 256 + t;          // float4 index into W
        const int k  = q >> 4;               // row of W (K index)
        const int n0 = (q & 15) << 2;        // first of 4 consecutive N cols
        const float4 w = ((const float4*)W)[q];
        Wt[(n0 + 0) * WT_STRIDE + k] = w.x;
        Wt[(n0 + 1) * WT_STRIDE + k] = w.y;
        Wt[(n0 + 2) * WT_STRIDE + k] = w.z;
        Wt[(n0 + 3) * WT_STRIDE + k] = w.w;
    }
}

// ---------------------------------------------------------------------------
// Kernel 1: h = x @ W   (f32 WMMA 16x16x4, transposed W in LDS)
// Each wave computes a 16-row x 64-col tile; 8 waves (256 thr) per block.
// ---------------------------------------------------------------------------
__global__ __launch_bounds__(256) void gemm_xw(const float* __restrict__ x,
                                               const float* __restrict__ W,
                                               float* __restrict__ h) {
    extern __shared__ float Wt[];           // [C_OUT][WT_STRIDE]
    const int t = threadIdx.x;
    load_W_transposed(W, Wt, t);
    __syncthreads();

    const int wave     = t >> 5;
    const int lane     = t & 31;
    const int halflane = lane >> 4;         // 0 or 1
    const int lid      = lane & 15;

    const int rowTile = blockIdx.x * 8 + wave;
    const int row0    = rowTile * 16;
    if (row0 >= N_NODES) return;

    int arow = row0 + lid;
    if (arow >= N_NODES) arow = N_NODES - 1;   // clamp loads, guard stores
    const float* xrow = x + (size_t)arow * D_IN;

    // Per-lane base pointers into transposed W for the 4 N-tiles.
    const float* wt0 = Wt + (size_t)(lid +  0) * WT_STRIDE;
    const float* wt1 = Wt + (size_t)(lid + 16) * WT_STRIDE;
    const float* wt2 = Wt + (size_t)(lid + 32) * WT_STRIDE;
    const float* wt3 = Wt + (size_t)(lid + 48) * WT_STRIDE;

    v8f acc0, acc1, acc2, acc3;
    const v8f vzero = {0.f,0.f,0.f,0.f,0.f,0.f,0.f,0.f};
    acc0 = acc1 = acc2 = acc3 = vzero;

    for (int k = 0; k < D_IN; k += 4) {
        const int ka = k + halflane * 2;
        v2f a  = *(const v2f*)(xrow + ka);   // A: 16x4, lane=M, halfwave=K pair
        v2f b0 = *(const v2f*)(wt0 + ka);    // single ds_load_b64 each
        v2f b1 = *(const v2f*)(wt1 + ka);
        v2f b2 = *(const v2f*)(wt2 + ka);
        v2f b3 = *(const v2f*)(wt3 + ka);
        acc0 = __builtin_amdgcn_wmma_f32_16x16x4_f32(false, a, false, b0, (short)0, acc0, false, false);
        acc1 = __builtin_amdgcn_wmma_f32_16x16x4_f32(false, a, false, b1, (short)0, acc1, false, false);
        acc2 = __builtin_amdgcn_wmma_f32_16x16x4_f32(false, a, false, b2, (short)0, acc2, false, false);
        acc3 = __builtin_amdgcn_wmma_f32_16x16x4_f32(false, a, false, b3, (short)0, acc3, false, false);
    }

    // D layout: VGPR v -> row (v + 8*halflane), col = lid (+16*tile)
    #pragma unroll
    for (int v = 0; v < 8; ++v) {
        const int m = row0 + v + 8 * halflane;
        if (m < N_NODES) {
            float* hm = h + (size_t)m * C_OUT + lid;
            hm[ 0] = acc0[v];
            hm[16] = acc1[v];
            hm[32] = acc2[v];
            hm[48] = acc3[v];
        }
    }
}

// ---------------------------------------------------------------------------
// Kernel 2: h[node_id[r]] += x[node_id[r]] @ W_id   (gathered WMMA + atomics;
// node_id may contain duplicates -> atomicAdd matches .at[].add semantics)
// ---------------------------------------------------------------------------
__global__ __launch_bounds__(256) void gemm_id(const float* __restrict__ x,
                                               const float* __restrict__ Wid,
                                               const int*   __restrict__ node_id,
                                               float* __restrict__ h) {
    extern __shared__ float Wt[];
    const int t = threadIdx.x;
    load_W_transposed(Wid, Wt, t);
    __syncthreads();

    const int wave     = t >> 5;
    const int lane     = t & 31;
    const int halflane = lane >> 4;
    const int lid      = lane & 15;

    const int rowTile = blockIdx.x * 8 + wave;
    const int row0    = rowTile * 16;
    if (row0 >= NID_CNT) return;

    int r = row0 + lid;
    if (r >= NID_CNT) r = NID_CNT - 1;
    const float* xrow = x + (size_t)node_id[r] * D_IN;

    const float* wt0 = Wt + (size_t)(lid +  0) * WT_STRIDE;
    const float* wt1 = Wt + (size_t)(lid + 16) * WT_STRIDE;
    const float* wt2 = Wt + (size_t)(lid + 32) * WT_STRIDE;
    const float* wt3 = Wt + (size_t)(lid + 48) * WT_STRIDE;

    v8f acc0, acc1, acc2, acc3;
    const v8f vzero = {0.f,0.f,0.f,0.f,0.f,0.f,0.f,0.f};
    acc0 = acc1 = acc2 = acc3 = vzero;

    for (int k = 0; k < D_IN; k += 4) {
        const int ka = k + halflane * 2;
        v2f a  = *(const v2f*)(xrow + ka);
        v2f b0 = *(const v2f*)(wt0 + ka);
        v2f b1 = *(const v2f*)(wt1 + ka);
        v2f b2 = *(const v2f*)(wt2 + ka);
        v2f b3 = *(const v2f*)(wt3 + ka);
        acc0 = __builtin_amdgcn_wmma_f32_16x16x4_f32(false, a, false, b0, (short)0, acc0, false, false);
        acc1 = __builtin_amdgcn_wmma_f32_16x16x4_f32(false, a, false, b1, (short)0, acc1, false, false);
        acc2 = __builtin_amdgcn_wmma_f32_16x16x4_f32(false, a, false, b2, (short)0, acc2, false, false);
        acc3 = __builtin_amdgcn_wmma_f32_16x16x4_f32(false, a, false, b3, (short)0, acc3, false, false);
    }

    #pragma unroll
    for (int v = 0; v < 8; ++v) {
        const int rr = row0 + v + 8 * halflane;
        if (rr < NID_CNT) {
            const int node = node_id[rr];
            float* hm = h + (size_t)node * C_OUT + lid;
            atomicAdd(hm +  0, acc0[v]);
            atomicAdd(hm + 16, acc1[v]);
            atomicAdd(hm + 32, acc2[v]);
            atomicAdd(hm + 48, acc3[v]);
        }
    }
}

// ---------------------------------------------------------------------------
// Kernel 3: per-node attention scalars + softmax-state init
// ai[n] = h[n] . att[0:64],  aj[n] = h[n] . att[64:128]
// ---------------------------------------------------------------------------
__global__ __launch_bounds__(256) void node_scalars(const float* __restrict__ h,
                                                    const float* __restrict__ att,
                                                    float* __restrict__ ai,
                                                    float* __restrict__ aj,
                                                    float* __restrict__ segmax,
                                                    float* __restrict__ segsum) {
    const int n = blockIdx.x * 256 + threadIdx.x;
    if (n >= N_NODES) return;
    const float4* hv  = (const float4*)(h + (size_t)n * C_OUT);
    const float4* a1v = (const float4*)att;             // att_i (x_i part)
    const float4* a2v = (const float4*)(att + C_OUT);   // att_j (x_j part)
    float si = 0.f, sj = 0.f;
    #pragma unroll
    for (int i = 0; i < C_OUT / 4; ++i) {
        const float4 hh = hv[i];
        const float4 a1 = a1v[i];
        const float4 a2 = a2v[i];
        si += hh.x * a1.x + hh.y * a1.y + hh.z * a1.z + hh.w * a1.w;
        sj += hh.x * a2.x + hh.y * a2.y + hh.z * a2.z + hh.w * a2.w;
    }
    ai[n] = si;
    aj[n] = sj;
    segmax[n] = -__builtin_inff();
    segsum[n] = 0.0f;
}

// ---------------------------------------------------------------------------
// Kernel 4: zero d_out (it is poisoned by the harness)
// ---------------------------------------------------------------------------
__global__ __launch_bounds__(256) void zero_out(float4* __restrict__ out, int n4) {
    const int i = blockIdx.x * 256 + threadIdx.x;
    if (i < n4) out[i] = (float4){0.f, 0.f, 0.f, 0.f};
}

// ---------------------------------------------------------------------------
// Kernel 5: edge logits + segment max   (edges E_TOT = E + N self-loops)
// ---------------------------------------------------------------------------
__global__ __launch_bounds__(256) void edge_alpha(const int* __restrict__ ei,
                                                  const float* __restrict__ ai,
                                                  const float* __restrict__ aj,
                                                  float* __restrict__ alpha,
                                                  float* __restrict__ segmax) {
    const int e = blockIdx.x * 256 + threadIdx.x;
    if (e >= E_TOT) return;
    int s, d;
    if (e < N_EDGES) { s = ei[e]; d = ei[N_EDGES + e]; }
    else             { s = d = e - N_EDGES; }           // self-loop
    float a = ai[d] + aj[s];
    a = a > 0.0f ? a : NEG_SLOPE * a;                   // leaky_relu
    alpha[e] = a;
    atomicMaxF(&segmax[d], a);
}

// ---------------------------------------------------------------------------
// Kernel 6: exp(alpha - max) + segment sum
// ---------------------------------------------------------------------------
__global__ __launch_bounds__(256) void edge_exp(const int* __restrict__ ei,
                                                const float* __restrict__ segmax,
                                                float* __restrict__ alpha,
                                                float* __restrict__ segsum) {
    const int e = blockIdx.x * 256 + threadIdx.x;
    if (e >= E_TOT) return;
    int d;
    if (e < N_EDGES) d = ei[N_EDGES + e];
    else             d = e - N_EDGES;
    const float w = expf(alpha[e] - segmax[d]);
    alpha[e] = w;
    atomicAdd(&segsum[d], w);
}

// ---------------------------------------------------------------------------
// Kernel 7: out[dst] += (alpha/segsum[dst]) * h[src]
// One wave per edge; each lane handles 2 contiguous cols (float2 gather).
// ---------------------------------------------------------------------------
__global__ __launch_bounds__(256) void edge_aggregate(const int* __restrict__ ei,
                                                      const float* __restrict__ h,
                                                      const float* __restrict__ alpha,
                                                      const float* __restrict__ segsum,
                                                      float* __restrict__ out) {
    const int e = blockIdx.x * 8 + (threadIdx.x >> 5);  // 8 waves per block
    if (e >= E_TOT) return;
    const int c = (threadIdx.x & 31) * 2;
    int s, d;
    if (e < N_EDGES) { s = ei[e]; d = ei[N_EDGES + e]; }
    else             { s = d = e - N_EDGES; }
    const float coeff = alpha[e] / (segsum[d] + 1e-16f);
    const float2 hv = *(const float2*)(h + (size_t)s * C_OUT + c);
    float* o = out + (size_t)d * C_OUT + c;
    atomicAdd(o,     coeff * hv.x);
    atomicAdd(o + 1, coeff * hv.y);
}

// ---------------------------------------------------------------------------
extern "C" void kernel_launch(void* const* d_in, const int* in_sizes, int n_in,
                              void* d_out, int out_size, void* d_ws, size_t ws_size,
                              hipStream_t stream) {
    const float* x       = (const float*)d_in[0];   // [N, 256]
    const int*   ei      = (const int*)  d_in[1];   // [2, E] (row0 src, row1 dst)
    const int*   node_id = (const int*)  d_in[2];   // [NID]
    const float* W       = (const float*)d_in[3];   // [256, 64]
    const float* W_id    = (const float*)d_in[4];   // [256, 64]
    const float* att     = (const float*)d_in[5];   // [1,1,128]
    float*       out     = (float*)d_out;           // [N, 64]

    // workspace layout
    float* h      = (float*)d_ws;                   // N*64
    float* ai     = h      + (size_t)N_NODES * C_OUT;
    float* aj     = ai     + N_NODES;
    float* segmax = aj     + N_NODES;
    float* segsum = segmax + N_NODES;
    float* alpha  = segsum + N_NODES;               // E_TOT

    // 1) h = x @ W            (16-row tiles, 8 waves/block)
    {
        const int tiles  = (N_NODES + 15) / 16;     // 6250
        const int blocks = (tiles + 7) / 8;         // 782
        gemm_xw<<<blocks, 256, LDS_BYTES, stream>>>(x, W, h);
    }
    // 2) h[node_id] += x[node_id] @ W_id
    {
        const int tiles  = (NID_CNT + 15) / 16;     // 625
        const int blocks = (tiles + 7) / 8;         // 79
        gemm_id<<<blocks, 256, LDS_BYTES, stream>>>(x, W_id, node_id, h);
    }
    // 3) per-node scalars + softmax state init
    node_scalars<<<(N_NODES + 255) / 256, 256, 0, stream>>>(h, att, ai, aj, segmax, segsum);
    // 4) zero output accumulator
    {
        const int n4 = N_NODES * C_OUT / 4;
        zero_out<<<(n4 + 255) / 256, 256, 0, stream>>>((float4*)out, n4);
    }
    // 5) edge logits + segment max
    edge_alpha<<<(E_TOT + 255) / 256, 256, 0, stream>>>(ei, ai, aj, alpha, segmax);
    // 6) exp + segment sum
    edge_exp<<<(E_TOT + 255) / 256, 256, 0, stream>>>(ei, segmax, alpha, segsum);
    // 7) weighted scatter aggregation (wave per edge)
    edge_aggregate<<<(E_TOT + 7) / 8, 256, 0, stream>>>(ei, h, alpha, segsum, out);
}